// GCN_4475355922529
// MI455X (gfx1250) — compile-verified
//
#include <hip/hip_runtime.h>
#include <hip/hip_bf16.h>

typedef __attribute__((ext_vector_type(2))) float v2f;
typedef __attribute__((ext_vector_type(8))) float v8f;

#define D_FEAT 128
#define LDS_STRIDE 132            // pad: bank = (4*lane + k) % 64 -> conflict-free reads
#define ROWS_PER_BLOCK 128        // 8 x 16-row panels share one LDS-resident W

// ---------------------------------------------------------------------------
// Degree / normalization kernels
// ---------------------------------------------------------------------------
__global__ void gcn_init_deg(float* __restrict__ deg, int N) {
    int i = blockIdx.x * blockDim.x + threadIdx.x;
    if (i < N) deg[i] = 1.0f;   // self-loop contributes 1
}

__global__ void gcn_count_deg(const int* __restrict__ dst, float* __restrict__ deg, int E) {
    int i = blockIdx.x * blockDim.x + threadIdx.x;
    if (i < E) atomicAdd(&deg[dst[i]], 1.0f);
}

__global__ void gcn_rsqrt_deg(float* __restrict__ dis, int N) {
    int i = blockIdx.x * blockDim.x + threadIdx.x;
    if (i < N) dis[i] = rsqrtf(dis[i]);   // deg >= 1 always (self loop)
}

// ---------------------------------------------------------------------------
// GEMM: H[N,128] = X[N,128] @ W[128,128] via V_WMMA_F32_16X16X4_F32.
// Block = 128 output rows (8 panels of 16); 8 wave32 waves each own a 16-col
// tile. W is loaded once per block into LDS *transposed* (k-major per column)
// so each B fragment is one aligned ds_load_b64; A panels staged in LDS with
// padded stride for conflict-free fragment reads.
//
// WMMA f32 16x16x4 VGPR layouts (ISA 7.12.2):
//   A 16x4 : lanes 0-15 rows 0-15 carry K={0,1}; lanes 16-31 carry K={2,3}
//   B 4x16 : lanes 0-15 cols 0-15 carry K={0,1}; lanes 16-31 carry K={2,3}
//   C 16x16: VGPR j -> row j (lanes 0-15) / row j+8 (lanes 16-31), col = lane&15
// ---------------------------------------------------------------------------
__global__ void __launch_bounds__(256)
gcn_gemm128(const float* __restrict__ X, const float* __restrict__ W,
            float* __restrict__ H, int N) {
    __shared__ float Ws[D_FEAT * LDS_STRIDE];   // W transposed: Ws[col*132 + k]
    __shared__ float As[16 * LDS_STRIDE];       // A panel:      As[row*132 + k]

    const int tid  = threadIdx.x;
    const int base = blockIdx.x * ROWS_PER_BLOCK;

    // --- one-time cooperative load of W, transposed into LDS ---------------
    // coalesced float4 row reads; scattered scalar LDS writes (one-time cost)
    for (int i = tid; i < (D_FEAT * D_FEAT) / 4; i += 256) {
        const int k = i >> 5;          // W row (input feature)
        const int c = i & 31;          // group of 4 output columns
        const float4 w4 = ((const float4*)(W + (size_t)k * D_FEAT))[c];
        Ws[(4 * c + 0) * LDS_STRIDE + k] = w4.x;
        Ws[(4 * c + 1) * LDS_STRIDE + k] = w4.y;
        Ws[(4 * c + 2) * LDS_STRIDE + k] = w4.z;
        Ws[(4 * c + 3) * LDS_STRIDE + k] = w4.w;
    }

    const int wave = tid >> 5;          // 0..7 -> 16-col tile
    const int lane = tid & 31;
    const int half = lane >> 4;         // 0: K={0,1}, 1: K={2,3}
    const int l16  = lane & 15;
    const int col  = wave * 16 + l16;

    #pragma unroll 1
    for (int p = 0; p < ROWS_PER_BLOCK / 16; ++p) {
        const int r0 = base + p * 16;
        if (r0 >= N) break;             // uniform across block

        __syncthreads();                // Ws ready / previous panel reads done

        // stage 16x128 A panel (coalesced float4 global, 16B-aligned LDS)
        for (int i = tid; i < (16 * D_FEAT) / 4; i += 256) {
            const int row = i >> 5;
            const int c   = i & 31;
            if (r0 + row < N) {
                const float4 a4 = ((const float4*)(X + (size_t)(r0 + row) * D_FEAT))[c];
                *(float4*)&As[row * LDS_STRIDE + 4 * c] = a4;
            }
        }
        __syncthreads();

        v8f acc = {};
        #pragma unroll
        for (int k0 = 0; k0 < D_FEAT; k0 += 4) {
            const int ka = k0 + 2 * half;
            const v2f a = *(const v2f*)(As + l16 * LDS_STRIDE + ka);
            const v2f b = *(const v2f*)(Ws + col * LDS_STRIDE + ka);
            acc = __builtin_amdgcn_wmma_f32_16x16x4_f32(
                /*neg_a=*/false, a, /*neg_b=*/false, b,
                /*c_mod=*/(short)0, acc, /*reuse_a=*/false, /*reuse_b=*/false);
        }

        #pragma unroll
        for (int j = 0; j < 8; ++j) {
            const int row = r0 + j + 8 * half;
            if (row < N) H[(size_t)row * D_FEAT + col] = acc[j];
        }
    }
}

// ---------------------------------------------------------------------------
// out[n,:] = b[:] + dis[n]^2 * H[n,:]   (self-loop term; also initializes out)
// ---------------------------------------------------------------------------
__global__ void gcn_self_bias(const float* __restrict__ H, const float* __restrict__ dis,
                              const float* __restrict__ b, float* __restrict__ out, int N) {
    int i = blockIdx.x * blockDim.x + threadIdx.x;
    if (i < N * D_FEAT) {
        int n = i >> 7;
        int d = i & (D_FEAT - 1);
        float s = dis[n];
        out[i] = b[d] + s * s * H[i];
    }
}

// ---------------------------------------------------------------------------
// One wave per edge: coalesced 512B gather of H[src], scaled scatter-add
// into out[dst] (L2-resident float atomics).
// ---------------------------------------------------------------------------
__global__ void gcn_scatter_edges(const float* __restrict__ H,
                                  const int* __restrict__ src,
                                  const int* __restrict__ dst,
                                  const float* __restrict__ dis,
                                  float* __restrict__ out, int E) {
    int t = blockIdx.x * blockDim.x + threadIdx.x;
    int e = t >> 5;
    if (e >= E) return;
    int lane = threadIdx.x & 31;
    int s = src[e];
    int d = dst[e];
    float w = dis[s] * dis[d];
    const float4 hv = ((const float4*)(H + (size_t)s * D_FEAT))[lane];
    float* o = out + (size_t)d * D_FEAT + lane * 4;
    atomicAdd(o + 0, hv.x * w);
    atomicAdd(o + 1, hv.y * w);
    atomicAdd(o + 2, hv.z * w);
    atomicAdd(o + 3, hv.w * w);
}

// ---------------------------------------------------------------------------
// Orchestration
// ---------------------------------------------------------------------------
extern "C" void kernel_launch(void* const* d_in, const int* in_sizes, int n_in,
                              void* d_out, int out_size, void* d_ws, size_t ws_size,
                              hipStream_t stream) {
    const float* feat = (const float*)d_in[0];     // [N,128]
    const int*   ei   = (const int*)d_in[1];       // [2,E]
    const float* W0   = (const float*)d_in[2];
    const float* b0   = (const float*)d_in[3];
    const float* W1   = (const float*)d_in[4];
    const float* b1   = (const float*)d_in[5];
    const float* W2   = (const float*)d_in[6];
    const float* b2   = (const float*)d_in[7];

    const int N = in_sizes[0] / D_FEAT;
    const int E = in_sizes[1] / 2;
    const int* src = ei;
    const int* dst = ei + E;

    // workspace layout
    float* dis = (float*)d_ws;                               // [N] (deg -> dis in place)
    size_t off = ((size_t)N + 1023) & ~(size_t)1023;
    float* H   = dis + off;                                  // [N,128]
    float* X   = H + (size_t)N * D_FEAT;                     // [N,128]
    float* out = (float*)d_out;

    const int TB = 256;
    dim3 blkN((N + TB - 1) / TB);
    dim3 blkE((E + TB - 1) / TB);
    dim3 blkND(((size_t)N * D_FEAT + TB - 1) / TB);
    dim3 blkGemm((N + ROWS_PER_BLOCK - 1) / ROWS_PER_BLOCK);
    dim3 blkEdge(((size_t)E * 32 + TB - 1) / TB);

    // normalization: deg -> dis = rsqrt(deg)
    gcn_init_deg<<<blkN, TB, 0, stream>>>(dis, N);
    gcn_count_deg<<<blkE, TB, 0, stream>>>(dst, dis, E);
    gcn_rsqrt_deg<<<blkN, TB, 0, stream>>>(dis, N);

    // layer 0: feat -> X
    gcn_gemm128<<<blkGemm, TB, 0, stream>>>(feat, W0, H, N);
    gcn_self_bias<<<blkND, TB, 0, stream>>>(H, dis, b0, X, N);
    gcn_scatter_edges<<<blkEdge, TB, 0, stream>>>(H, src, dst, dis, X, E);

    // layer 1: X -> X (GEMM consumes X before aggregate rewrites it; stream-ordered)
    gcn_gemm128<<<blkGemm, TB, 0, stream>>>(X, W1, H, N);
    gcn_self_bias<<<blkND, TB, 0, stream>>>(H, dis, b1, X, N);
    gcn_scatter_edges<<<blkEdge, TB, 0, stream>>>(H, src, dst, dis, X, E);

    // layer 2: X -> d_out
    gcn_gemm128<<<blkGemm, TB, 0, stream>>>(X, W2, H, N);
    gcn_self_bias<<<blkND, TB, 0, stream>>>(H, dis, b2, out, N);
    gcn_scatter_edges<<<blkEdge, TB, 0, stream>>>(H, src, dst, dis, out, E);
}